// PointNetSetAbstraction_30056181138038
// MI455X (gfx1250) — compile-verified
//
#include <hip/hip_runtime.h>

#define B_   16
#define N_   4096
#define D_   64
#define S_   1024
#define NS_  32
#define C0_  67
#define RAD2 0.04f
#define BN_EPS_ 1e-5f

typedef __attribute__((ext_vector_type(2))) float v2f;
typedef __attribute__((ext_vector_type(8))) float v8f;

// D = A(16x4) * B(4x16) + C, full fp32 (matches reference numerics)
__device__ __forceinline__ v8f wmma4(v2f a, v2f b, v8f c) {
  return __builtin_amdgcn_wmma_f32_16x16x4_f32(false, a, false, b, (short)0, c,
                                               false, false);
}

// ---------------------------------------------------------------------------
// Kernel 1: farthest point sampling. One block per batch; 4 points per thread
// held in registers; per-step two-level (shuffle + LDS) argmax with
// first-index tie-break to match jnp.argmax.
// ---------------------------------------------------------------------------
__global__ __launch_bounds__(1024) void fps_kernel(const float* __restrict__ xyz,
                                                   int* __restrict__ fps_idx,
                                                   float* __restrict__ new_xyz) {
  int b = blockIdx.x;
  int t = threadIdx.x;
  int lane = t & 31, wv = t >> 5;
  const float* xb = xyz + (size_t)b * N_ * 3;
  float px[4], py[4], pz[4], dist[4];
#pragma unroll
  for (int j = 0; j < 4; ++j) {
    int p = t + j * 1024;
    px[j] = xb[p * 3 + 0];
    py[j] = xb[p * 3 + 1];
    pz[j] = xb[p * 3 + 2];
    dist[j] = 1e10f;
  }
  __shared__ float s_val[32];
  __shared__ int   s_idx[32];
  __shared__ float s_c[3];
  __shared__ int   s_cur;
  int cur = 0;
  for (int i = 0; i < S_; ++i) {
    if (t == (cur & 1023)) {                // owner broadcasts centroid coords
      int j = cur >> 10;
      s_c[0] = px[j]; s_c[1] = py[j]; s_c[2] = pz[j];
    }
    __syncthreads();
    float cx = s_c[0], cy = s_c[1], cz = s_c[2];
    if (t == 0) {
      fps_idx[b * S_ + i] = cur;
      float* o = new_xyz + ((size_t)b * S_ + i) * 3;
      o[0] = cx; o[1] = cy; o[2] = cz;
    }
    float bv = -1.0f; int bi = 0x7fffffff;
#pragma unroll
    for (int j = 0; j < 4; ++j) {
      float dx = px[j] - cx, dy = py[j] - cy, dz = pz[j] - cz;
      float d = dx * dx + dy * dy + dz * dz;
      dist[j] = fminf(dist[j], d);
      int p = t + j * 1024;
      if (dist[j] > bv || (dist[j] == bv && p < bi)) { bv = dist[j]; bi = p; }
    }
#pragma unroll
    for (int m = 16; m >= 1; m >>= 1) {
      float ov = __shfl_xor(bv, m, 32);
      int   oi = __shfl_xor(bi, m, 32);
      if (ov > bv || (ov == bv && oi < bi)) { bv = ov; bi = oi; }
    }
    if (lane == 0) { s_val[wv] = bv; s_idx[wv] = bi; }
    __syncthreads();
    if (wv == 0) {
      bv = s_val[lane]; bi = s_idx[lane];
#pragma unroll
      for (int m = 16; m >= 1; m >>= 1) {
        float ov = __shfl_xor(bv, m, 32);
        int   oi = __shfl_xor(bi, m, 32);
        if (ov > bv || (ov == bv && oi < bi)) { bv = ov; bi = oi; }
      }
      if (lane == 0) s_cur = bi;
    }
    __syncthreads();
    cur = s_cur;
    __syncthreads();
  }
}

// ---------------------------------------------------------------------------
// Kernel 2: ball query = 32 smallest d^2 (top_k semantics, stable ties),
// radius mask replaces with nearest index. One wave per centroid, d^2 in LDS.
// ---------------------------------------------------------------------------
__global__ __launch_bounds__(64) void ballq_kernel(const float* __restrict__ xyz,
                                                   const float* __restrict__ new_xyz,
                                                   int* __restrict__ idx_ball) {
  __shared__ float d2s[2][N_];
  int lane = threadIdx.x & 31, wv = threadIdx.x >> 5;
  int gs = blockIdx.x * 2 + wv;            // b*S + s
  int b = gs >> 10;
  const float* xb = xyz + (size_t)b * N_ * 3;
  float cx = new_xyz[gs * 3 + 0], cy = new_xyz[gs * 3 + 1], cz = new_xyz[gs * 3 + 2];
  float* my = d2s[wv];
  for (int j = 0; j < N_ / 32; ++j) {
    int p = lane + j * 32;
    float dx = xb[p * 3 + 0] - cx, dy = xb[p * 3 + 1] - cy, dz = xb[p * 3 + 2] - cz;
    my[p] = dx * dx + dy * dy + dz * dz;
  }
  __syncthreads();
  int first = 0;
  int* out = idx_ball + (size_t)gs * NS_;
  for (int r = 0; r < NS_; ++r) {
    float bv = 3.4e38f; int bi = 0x7fffffff;
    for (int j = 0; j < N_ / 32; ++j) {
      int p = lane + j * 32;
      float v = my[p];
      if (v < bv || (v == bv && p < bi)) { bv = v; bi = p; }
    }
#pragma unroll
    for (int m = 16; m >= 1; m >>= 1) {
      float ov = __shfl_xor(bv, m, 32);
      int   oi = __shfl_xor(bi, m, 32);
      if (ov < bv || (ov == bv && oi < bi)) { bv = ov; bi = oi; }
    }
    if (r == 0) first = bi;
    int sel = (bv > RAD2) ? first : bi;
    if (lane == 0) { out[r] = sel; my[bi] = 3.4e38f; }
    __syncthreads();
  }
}

// ---------------------------------------------------------------------------
// WMMA GEMM tile helpers. One wave owns one centroid tile (N = 32 samples).
// A = weights [M][KACT] row-major from global (L2-hot), B = activations in
// LDS as [K][32]. Output = two 16x16 C tiles (cols 0..15 / 16..31).
// A layout:  lanes 0-15 hold K=0,1 ; lanes 16-31 hold K=2,3 (per k-step)
// C layout:  vgpr v -> row mt*16 + v + 8*(lane>=16), col = lane%16
// ---------------------------------------------------------------------------
template <int KSTEPS, int KACT>
__device__ __forceinline__ void gemm_mtile(const float* __restrict__ Wg,
                                           const float* __restrict__ bias,
                                           const float (*__restrict__ xin)[32],
                                           int mt, int lane, v8f& c0, v8f& c1) {
  int half = lane >> 4, lq = lane & 15;
  v8f a0{}, a1{};
#pragma unroll
  for (int v = 0; v < 8; ++v) {
    float bb = bias[mt * 16 + 8 * half + v];
    a0[v] = bb; a1[v] = bb;
  }
  c0 = a0; c1 = a1;
#pragma unroll
  for (int ks = 0; ks < KSTEPS; ++ks) {
    int kc = ks * 4 + 2 * half;
    int m = mt * 16 + lq;
    v2f a, b0, b1;
    a.x = (kc < KACT)     ? Wg[m * KACT + kc]     : 0.0f;
    a.y = (kc + 1 < KACT) ? Wg[m * KACT + kc + 1] : 0.0f;
    b0.x = xin[kc][lq];        b0.y = xin[kc + 1][lq];
    b1.x = xin[kc][16 + lq];   b1.y = xin[kc + 1][16 + lq];
    c0 = wmma4(a, b0, c0);
    c1 = wmma4(a, b1, c1);
  }
}

__device__ __forceinline__ void accum_stats(const v8f& c0, const v8f& c1, int mt,
                                            int lane, float* lsum, float* lsq) {
  int half = lane >> 4, lq = lane & 15;
#pragma unroll
  for (int v = 0; v < 8; ++v) {
    float s = c0[v] + c1[v];
    float q = c0[v] * c0[v] + c1[v] * c1[v];
#pragma unroll
    for (int m = 1; m < 16; m <<= 1) {       // reduce over the 16 cols in group
      s += __shfl_xor(s, m, 32);
      q += __shfl_xor(q, m, 32);
    }
    if (lq == 0) {
      int row = mt * 16 + 8 * half + v;
      atomicAdd(&lsum[row], s);
      atomicAdd(&lsq[row], q);
    }
  }
}

__device__ __forceinline__ void bn_store(const v8f& c0, const v8f& c1, int mt,
                                         int lane, const float* __restrict__ sc,
                                         const float* __restrict__ sh,
                                         float (*__restrict__ xout)[32]) {
  int half = lane >> 4, lq = lane & 15;
#pragma unroll
  for (int v = 0; v < 8; ++v) {
    int row = mt * 16 + 8 * half + v;
    float s = sc[row], h = sh[row];
    xout[row][lq]      = fmaxf(c0[v] * s + h, 0.0f);
    xout[row][16 + lq] = fmaxf(c1[v] * s + h, 0.0f);
  }
}

__device__ __forceinline__ void bn_max_out(const v8f& c0, const v8f& c1, int mt,
                                           int lane, const float* __restrict__ sc,
                                           const float* __restrict__ sh,
                                           float* __restrict__ outp) {
  int half = lane >> 4, lq = lane & 15;
#pragma unroll
  for (int v = 0; v < 8; ++v) {
    int row = mt * 16 + 8 * half + v;
    float s = sc[row], h = sh[row];
    float y = fmaxf(fmaxf(c0[v] * s + h, 0.0f), fmaxf(c1[v] * s + h, 0.0f));
#pragma unroll
    for (int m = 1; m < 16; m <<= 1) y = fmaxf(y, __shfl_xor(y, m, 32));
    if (lq == 0) outp[row] = y;
  }
}

// Stage gathered 67-channel input tile into LDS [68][32] (row 67 zero pad).
__device__ __forceinline__ void stage_x0(const float* __restrict__ xyz,
                                         const float* __restrict__ points,
                                         const float* __restrict__ new_xyz,
                                         const int* __restrict__ idx_ball,
                                         int gs, int lane,
                                         float (*__restrict__ x0)[32]) {
  int b = gs >> 10;
  int pi = idx_ball[(size_t)gs * NS_ + lane];
  float cx = new_xyz[gs * 3 + 0], cy = new_xyz[gs * 3 + 1], cz = new_xyz[gs * 3 + 2];
  const float* xp = xyz + ((size_t)b * N_ + pi) * 3;
  x0[0][lane] = xp[0] - cx;
  x0[1][lane] = xp[1] - cy;
  x0[2][lane] = xp[2] - cz;
  const float* pp = points + ((size_t)b * N_ + pi) * D_;
#pragma unroll
  for (int c = 0; c < D_; c += 4) {
    float4 v = *(const float4*)(pp + c);
    x0[3 + c][lane] = v.x; x0[4 + c][lane] = v.y;
    x0[5 + c][lane] = v.z; x0[6 + c][lane] = v.w;
  }
  x0[67][lane] = 0.0f;
}

// ---------------------------------------------------------------------------
// MLP phase kernel. PHASE 0/1/2: recompute chain up to layer P, accumulate
// pre-BN channel stats of layer P. PHASE 3: full chain + relu + max + output.
// Two waves per block, one centroid per wave.
// ---------------------------------------------------------------------------
template <int PHASE>
__global__ __launch_bounds__(64) void mlp_kernel(
    const float* __restrict__ xyz, const float* __restrict__ points,
    const float* __restrict__ W0, const float* __restrict__ b0,
    const float* __restrict__ W1, const float* __restrict__ b1,
    const float* __restrict__ W2, const float* __restrict__ b2,
    const float* __restrict__ new_xyz, const int* __restrict__ idx_ball,
    const float* __restrict__ sc0, const float* __restrict__ sh0,
    const float* __restrict__ sc1, const float* __restrict__ sh1,
    const float* __restrict__ sc2, const float* __restrict__ sh2,
    float* __restrict__ st_sum, float* __restrict__ st_sq,
    float* __restrict__ out_points) {
  __shared__ float x0s[2][68][32];
  __shared__ float x1s[2][64][32];
  __shared__ float lsum[128], lsq[128];
  int lane = threadIdx.x & 31, wv = threadIdx.x >> 5;
  if (PHASE != 3) {
    for (int i = threadIdx.x; i < 128; i += 64) { lsum[i] = 0.0f; lsq[i] = 0.0f; }
    __syncthreads();
  }
  int gs = blockIdx.x * 2 + wv;
  float (*x0)[32] = x0s[wv];
  float (*x1)[32] = x1s[wv];

  stage_x0(xyz, points, new_xyz, idx_ball, gs, lane, x0);

  v8f c0, c1;
  // ---- layer 0: 67 -> 64
#pragma unroll
  for (int mt = 0; mt < 4; ++mt) {
    gemm_mtile<17, C0_>(W0, b0, x0, mt, lane, c0, c1);
    if (PHASE == 0) accum_stats(c0, c1, mt, lane, lsum, lsq);
    else            bn_store(c0, c1, mt, lane, sc0, sh0, x1);
  }
  if (PHASE >= 1) {
    // ---- layer 1: 64 -> 64  (reads x1, writes bn-relu result into x0 rows 0..63)
#pragma unroll
    for (int mt = 0; mt < 4; ++mt) {
      gemm_mtile<16, 64>(W1, b1, x1, mt, lane, c0, c1);
      if (PHASE == 1) accum_stats(c0, c1, mt, lane, lsum, lsq);
      else            bn_store(c0, c1, mt, lane, sc1, sh1, x0);
    }
  }
  if (PHASE >= 2) {
    // ---- layer 2: 64 -> 128
#pragma unroll
    for (int mt = 0; mt < 8; ++mt) {
      gemm_mtile<16, 64>(W2, b2, x0, mt, lane, c0, c1);
      if (PHASE == 2) accum_stats(c0, c1, mt, lane, lsum, lsq);
      else            bn_max_out(c0, c1, mt, lane, sc2, sh2,
                                 out_points + (size_t)gs * 128);
    }
  }
  if (PHASE != 3) {
    __syncthreads();
    int nch = (PHASE == 2) ? 128 : 64;
    for (int ch = threadIdx.x; ch < nch; ch += 64) {
      atomicAdd(&st_sum[ch], lsum[ch]);
      atomicAdd(&st_sq[ch], lsq[ch]);
    }
  }
}

__global__ void bn_finalize_kernel(const float* __restrict__ sum,
                                   const float* __restrict__ sq,
                                   const float* __restrict__ gamma,
                                   const float* __restrict__ beta,
                                   float* __restrict__ scale,
                                   float* __restrict__ shift, int nch) {
  int c = blockIdx.x * blockDim.x + threadIdx.x;
  if (c >= nch) return;
  const float M = (float)B_ * S_ * NS_;
  float mean = sum[c] / M;
  float var = sq[c] / M - mean * mean;
  float sc = gamma[c] * rsqrtf(var + BN_EPS_);
  scale[c] = sc;
  shift[c] = beta[c] - mean * sc;
}

__global__ void zero_kernel(float* __restrict__ p, int n) {
  int i = blockIdx.x * blockDim.x + threadIdx.x;
  if (i < n) p[i] = 0.0f;
}

// ---------------------------------------------------------------------------
extern "C" void kernel_launch(void* const* d_in, const int* in_sizes, int n_in,
                              void* d_out, int out_size, void* d_ws, size_t ws_size,
                              hipStream_t stream) {
  const float* xyz    = (const float*)d_in[0];
  const float* points = (const float*)d_in[1];
  const float* W0 = (const float*)d_in[2];
  const float* b0 = (const float*)d_in[3];
  const float* g0 = (const float*)d_in[4];
  const float* be0 = (const float*)d_in[5];
  const float* W1 = (const float*)d_in[6];
  const float* b1 = (const float*)d_in[7];
  const float* g1 = (const float*)d_in[8];
  const float* be1 = (const float*)d_in[9];
  const float* W2 = (const float*)d_in[10];
  const float* b2 = (const float*)d_in[11];
  const float* g2 = (const float*)d_in[12];
  const float* be2 = (const float*)d_in[13];

  float* out = (float*)d_out;
  float* new_xyz = out;                               // (B, S, 3)
  float* new_points = out + (size_t)B_ * S_ * 3;      // (B, S, 128)

  char* ws = (char*)d_ws;
  int* fps_idx  = (int*)ws;                                       // 64 KB
  int* idx_ball = (int*)(ws + 65536);                             // 2 MB
  float* stats = (float*)(ws + 65536 + (size_t)B_ * S_ * NS_ * 4);
  float* sum_b = stats;          // 3 * 128
  float* sq_b  = stats + 384;    // 3 * 128
  float* sc_b  = stats + 768;    // 3 * 128
  float* sh_b  = stats + 1152;   // 3 * 128

  fps_kernel<<<B_, 1024, 0, stream>>>(xyz, fps_idx, new_xyz);
  ballq_kernel<<<(B_ * S_) / 2, 64, 0, stream>>>(xyz, new_xyz, idx_ball);
  zero_kernel<<<3, 256, 0, stream>>>(stats, 768);

  mlp_kernel<0><<<(B_ * S_) / 2, 64, 0, stream>>>(
      xyz, points, W0, b0, W1, b1, W2, b2, new_xyz, idx_ball,
      sc_b, sh_b, sc_b + 128, sh_b + 128, sc_b + 256, sh_b + 256,
      sum_b, sq_b, new_points);
  bn_finalize_kernel<<<1, 64, 0, stream>>>(sum_b, sq_b, g0, be0, sc_b, sh_b, 64);

  mlp_kernel<1><<<(B_ * S_) / 2, 64, 0, stream>>>(
      xyz, points, W0, b0, W1, b1, W2, b2, new_xyz, idx_ball,
      sc_b, sh_b, sc_b + 128, sh_b + 128, sc_b + 256, sh_b + 256,
      sum_b + 128, sq_b + 128, new_points);
  bn_finalize_kernel<<<1, 64, 0, stream>>>(sum_b + 128, sq_b + 128, g1, be1,
                                           sc_b + 128, sh_b + 128, 64);

  mlp_kernel<2><<<(B_ * S_) / 2, 64, 0, stream>>>(
      xyz, points, W0, b0, W1, b1, W2, b2, new_xyz, idx_ball,
      sc_b, sh_b, sc_b + 128, sh_b + 128, sc_b + 256, sh_b + 256,
      sum_b + 256, sq_b + 256, new_points);
  bn_finalize_kernel<<<1, 128, 0, stream>>>(sum_b + 256, sq_b + 256, g2, be2,
                                            sc_b + 256, sh_b + 256, 128);

  mlp_kernel<3><<<(B_ * S_) / 2, 64, 0, stream>>>(
      xyz, points, W0, b0, W1, b1, W2, b2, new_xyz, idx_ball,
      sc_b, sh_b, sc_b + 128, sh_b + 128, sc_b + 256, sh_b + 256,
      sum_b, sq_b, new_points);
}